// SAGE_858993459677
// MI455X (gfx1250) — compile-verified
//
#include <hip/hip_runtime.h>
#include <hip/hip_bf16.h>

#define N_NODES  50000
#define N_EDGES  800000
#define N_GRAPHS 1000
#define X_DIM    128
#define HIDDEN   256
#define N_LAYERS 3
#define N_CLASS  10

typedef __attribute__((ext_vector_type(2))) float v2f;
typedef __attribute__((ext_vector_type(8))) float v8f;

// ---------------------------------------------------------------------------
// fp32 WMMA helper: D = A(16x4) * B(4x16) + C   (V_WMMA_F32_16X16X4_F32)
// ---------------------------------------------------------------------------
__device__ __forceinline__ v8f wmma4_f32(v2f a, v2f b, v8f c) {
    return __builtin_amdgcn_wmma_f32_16x16x4_f32(
        /*neg_a=*/false, a, /*neg_b=*/false, b,
        /*c_mod=*/(short)0, c, /*reuse_a=*/false, /*reuse_b=*/false);
}

__device__ __forceinline__ void atomic_add_f32(float* p, float v) {
    __hip_atomic_fetch_add(p, v, __ATOMIC_RELAXED, __HIP_MEMORY_SCOPE_AGENT);
}

// ---------------------------------------------------------------------------
// Pack a row-major KxN (N==256) weight matrix into WMMA B-fragment order:
//   P[ ((kg*16 + ct)*32 + lane)*2 + j ] = W[4*kg + 2*(lane>>4) + j][16*ct + (lane&15)]
// so a wave's B fragment for (k-group kg, col-tile ct) is one coalesced b64 load.
// total threads = Krows * 256
// ---------------------------------------------------------------------------
__global__ void pack_w_kernel(const float* __restrict__ W, float* __restrict__ P,
                              int Krows) {
    const int t = blockIdx.x * blockDim.x + threadIdx.x;
    if (t >= Krows * HIDDEN) return;
    const int j    = t & 1;
    const int lane = (t >> 1) & 31;
    const int ct   = (t >> 6) & 15;
    const int kg   = t >> 10;
    const int lo = lane & 15;
    const int hi = lane >> 4;
    const int k = 4 * kg + 2 * hi + j;
    const int n = 16 * ct + lo;
    P[t] = W[(size_t)k * HIDDEN + n];
}

// ---------------------------------------------------------------------------
// degree / inverse degree
// ---------------------------------------------------------------------------
__global__ void degree_kernel(const int* __restrict__ dst, float* __restrict__ deg) {
    int e = blockIdx.x * blockDim.x + threadIdx.x;
    if (e < N_EDGES) atomic_add_f32(&deg[dst[e]], 1.0f);
}

__global__ void invdeg_kernel(float* __restrict__ deg) {
    int i = blockIdx.x * blockDim.x + threadIdx.x;
    if (i < N_NODES) deg[i] = 1.0f / fmaxf(deg[i], 1.0f);
}

// ---------------------------------------------------------------------------
// Encoder: H = X[50000,128] @ W[128,256] + bias   (WMMA f32 16x16x4)
// grid.x = 3125 (M tiles of 16), block = 128 (4 waves; wave w owns N cols w*64..+63)
// Pw = packed W_enc fragments.
// ---------------------------------------------------------------------------
__global__ void encoder_gemm(const float* __restrict__ X, const float* __restrict__ Pw,
                             const float* __restrict__ bias, float* __restrict__ H) {
    const int lane = threadIdx.x & 31;
    const int wave = threadIdx.x >> 5;
    const int lo = lane & 15;
    const int hi = lane >> 4;
    const int m0 = blockIdx.x * 16;
    const int ct0 = wave * 4;                 // first col-tile of this wave

    v8f c[4];
#pragma unroll
    for (int t = 0; t < 4; ++t) {
        const float bv = bias[(ct0 + t) * 16 + lo];
#pragma unroll
        for (int r = 0; r < 8; ++r) c[t][r] = bv;
    }

    const float* arow = X + (size_t)(m0 + lo) * X_DIM + 2 * hi;
#pragma unroll 2
    for (int kg = 0; kg < X_DIM / 4; ++kg) {
        const v2f a = *(const v2f*)(arow + 4 * kg);
#pragma unroll
        for (int t = 0; t < 4; ++t) {
            const v2f b = *(const v2f*)(Pw + (((size_t)kg * 16 + ct0 + t) * 32 + lane) * 2);
            c[t] = wmma4_f32(a, b, c[t]);
        }
    }

#pragma unroll
    for (int t = 0; t < 4; ++t)
#pragma unroll
        for (int r = 0; r < 8; ++r)
            H[(size_t)(m0 + r + 8 * hi) * HIDDEN + (ct0 + t) * 16 + lo] = c[t][r];
}

// ---------------------------------------------------------------------------
// Edge scatter: agg[dst] += h[src] * atten   (64 threads per edge, float4 each)
// ---------------------------------------------------------------------------
__global__ void scatter_edges(const float* __restrict__ H, const int* __restrict__ src,
                              const int* __restrict__ dst, const float* __restrict__ atten,
                              float* __restrict__ agg) {
    const long long t = (long long)blockIdx.x * blockDim.x + threadIdx.x;
    const int e = (int)(t >> 6);
    const int q = (int)(t & 63);
    if (e >= N_EDGES) return;
    const int   s = src[e];
    const int   d = dst[e];
    const float a = atten[e];
    const float4 hv = *(const float4*)(H + (size_t)s * HIDDEN + q * 4);
    float* p = agg + (size_t)d * HIDDEN + q * 4;
    atomic_add_f32(p + 0, hv.x * a);
    atomic_add_f32(p + 1, hv.y * a);
    atomic_add_f32(p + 2, hv.z * a);
    atomic_add_f32(p + 3, hv.w * a);
}

// ---------------------------------------------------------------------------
// SAGE combine: Hout = relu( (agg*inv_deg) @ Wl + bl + Hin @ Wr )
// Both 256-K matmuls fused into one accumulator. grid.x = 3125, block = 128.
// Pl/Pr = packed fragment weights.
// ---------------------------------------------------------------------------
__global__ void combine_gemm(const float* __restrict__ AGG, const float* __restrict__ invd,
                             const float* __restrict__ Hin,
                             const float* __restrict__ Pl, const float* __restrict__ bl,
                             const float* __restrict__ Pr, float* __restrict__ Hout) {
    const int lane = threadIdx.x & 31;
    const int wave = threadIdx.x >> 5;
    const int lo = lane & 15;
    const int hi = lane >> 4;
    const int m0 = blockIdx.x * 16;
    const int ct0 = wave * 4;

    v8f c[4];
#pragma unroll
    for (int t = 0; t < 4; ++t) {
        const float bv = bl[(ct0 + t) * 16 + lo];
#pragma unroll
        for (int r = 0; r < 8; ++r) c[t][r] = bv;
    }

    const float scale = invd[m0 + lo];                    // row scale for agg rows
    const float* aggrow = AGG + (size_t)(m0 + lo) * HIDDEN + 2 * hi;
    const float* hrow   = Hin + (size_t)(m0 + lo) * HIDDEN + 2 * hi;

#pragma unroll 2
    for (int kg = 0; kg < HIDDEN / 4; ++kg) {
        v2f a1 = *(const v2f*)(aggrow + 4 * kg);
        a1.x *= scale;
        a1.y *= scale;
        const v2f a2 = *(const v2f*)(hrow + 4 * kg);
#pragma unroll
        for (int t = 0; t < 4; ++t) {
            const size_t fidx = (((size_t)kg * 16 + ct0 + t) * 32 + lane) * 2;
            const v2f b1 = *(const v2f*)(Pl + fidx);
            const v2f b2 = *(const v2f*)(Pr + fidx);
            c[t] = wmma4_f32(a1, b1, c[t]);
            c[t] = wmma4_f32(a2, b2, c[t]);
        }
    }

#pragma unroll
    for (int t = 0; t < 4; ++t)
#pragma unroll
        for (int r = 0; r < 8; ++r)
            Hout[(size_t)(m0 + r + 8 * hi) * HIDDEN + (ct0 + t) * 16 + lo] =
                fmaxf(c[t][r], 0.0f);
}

// ---------------------------------------------------------------------------
// Global add-pool: pooled[batch[n]] += h[n]   (64 threads per node, float4 each)
// ---------------------------------------------------------------------------
__global__ void pool_kernel(const float* __restrict__ H, const int* __restrict__ batch,
                            float* __restrict__ pooled) {
    const long long t = (long long)blockIdx.x * blockDim.x + threadIdx.x;
    const int n = (int)(t >> 6);
    const int q = (int)(t & 63);
    if (n >= N_NODES) return;
    const int g = batch[n];
    const float4 hv = *(const float4*)(H + (size_t)n * HIDDEN + q * 4);
    float* p = pooled + (size_t)g * HIDDEN + q * 4;
    atomic_add_f32(p + 0, hv.x);
    atomic_add_f32(p + 1, hv.y);
    atomic_add_f32(p + 2, hv.z);
    atomic_add_f32(p + 3, hv.w);
}

// ---------------------------------------------------------------------------
// Head: out[1000,10] = pooled @ W_out + b_out  (tiny; one thread per output)
// ---------------------------------------------------------------------------
__global__ void head_kernel(const float* __restrict__ pooled, const float* __restrict__ W,
                            const float* __restrict__ b, float* __restrict__ out) {
    const int t = blockIdx.x * blockDim.x + threadIdx.x;
    if (t >= N_GRAPHS * N_CLASS) return;
    const int g = t / N_CLASS;
    const int c = t % N_CLASS;
    float acc = b[c];
    const float* pr = pooled + (size_t)g * HIDDEN;
    for (int k = 0; k < HIDDEN; ++k) acc += pr[k] * W[k * N_CLASS + c];
    out[t] = acc;
}

// ---------------------------------------------------------------------------
extern "C" void kernel_launch(void* const* d_in, const int* in_sizes, int n_in,
                              void* d_out, int out_size, void* d_ws, size_t ws_size,
                              hipStream_t stream) {
    const float* x     = (const float*)d_in[0];
    const int*   edge  = (const int*)  d_in[1];
    const int*   batch = (const int*)  d_in[2];
    const float* atten = (const float*)d_in[3];
    const float* W_enc = (const float*)d_in[4];
    const float* b_enc = (const float*)d_in[5];
    const float* W_l   = (const float*)d_in[6];
    const float* b_l   = (const float*)d_in[7];
    const float* W_r   = (const float*)d_in[8];
    const float* W_out = (const float*)d_in[9];
    const float* b_out = (const float*)d_in[10];

    const int* src = edge;            // edge_index[0]
    const int* dst = edge + N_EDGES;  // edge_index[1]

    const size_t nodeFeatBytes = (size_t)N_NODES * HIDDEN * sizeof(float);
    char* ws = (char*)d_ws;
    float* hA     = (float*)ws; ws += nodeFeatBytes;
    float* hB     = (float*)ws; ws += nodeFeatBytes;
    float* agg    = (float*)ws; ws += nodeFeatBytes;
    float* invd   = (float*)ws; ws += (size_t)N_NODES * sizeof(float);
    float* pooled = (float*)ws; ws += (size_t)N_GRAPHS * HIDDEN * sizeof(float);
    float* Penc   = (float*)ws; ws += (size_t)X_DIM * HIDDEN * sizeof(float);
    float* Pl     = (float*)ws; ws += (size_t)N_LAYERS * HIDDEN * HIDDEN * sizeof(float);
    float* Pr     = (float*)ws; ws += (size_t)N_LAYERS * HIDDEN * HIDDEN * sizeof(float);
    (void)ws_size;

    // 0) repack weights into WMMA fragment order
    pack_w_kernel<<<(X_DIM * HIDDEN + 255) / 256, 256, 0, stream>>>(W_enc, Penc, X_DIM);
    for (int l = 0; l < N_LAYERS; ++l) {
        pack_w_kernel<<<(HIDDEN * HIDDEN + 255) / 256, 256, 0, stream>>>(
            W_l + (size_t)l * HIDDEN * HIDDEN, Pl + (size_t)l * HIDDEN * HIDDEN, HIDDEN);
        pack_w_kernel<<<(HIDDEN * HIDDEN + 255) / 256, 256, 0, stream>>>(
            W_r + (size_t)l * HIDDEN * HIDDEN, Pr + (size_t)l * HIDDEN * HIDDEN, HIDDEN);
    }

    // 1) degree -> inverse degree
    hipMemsetAsync(invd, 0, (size_t)N_NODES * sizeof(float), stream);
    degree_kernel<<<(N_EDGES + 255) / 256, 256, 0, stream>>>(dst, invd);
    invdeg_kernel<<<(N_NODES + 255) / 256, 256, 0, stream>>>(invd);

    // 2) node encoder
    encoder_gemm<<<N_NODES / 16, 128, 0, stream>>>(x, Penc, b_enc, hA);

    // 3) SAGE layers (ping-pong hA/hB)
    float* hc = hA;
    float* hn = hB;
    const long long scatterThreads = (long long)N_EDGES * 64;
    for (int l = 0; l < N_LAYERS; ++l) {
        hipMemsetAsync(agg, 0, nodeFeatBytes, stream);
        scatter_edges<<<(unsigned)(scatterThreads / 256), 256, 0, stream>>>(
            hc, src, dst, atten, agg);
        combine_gemm<<<N_NODES / 16, 128, 0, stream>>>(
            agg, invd, hc,
            Pl + (size_t)l * HIDDEN * HIDDEN, b_l + (size_t)l * HIDDEN,
            Pr + (size_t)l * HIDDEN * HIDDEN, hn);
        float* tmp = hc; hc = hn; hn = tmp;
    }

    // 4) global add-pool + head
    hipMemsetAsync(pooled, 0, (size_t)N_GRAPHS * HIDDEN * sizeof(float), stream);
    const long long poolThreads = (long long)N_NODES * 64;
    pool_kernel<<<(unsigned)(poolThreads / 256), 256, 0, stream>>>(hc, batch, pooled);
    head_kernel<<<(N_GRAPHS * N_CLASS + 127) / 128, 128, 0, stream>>>(
        pooled, W_out, b_out, (float*)d_out);
}